// MultiHeadAttention_6408091206170
// MI455X (gfx1250) — compile-verified
//
#include <hip/hip_runtime.h>
#include <math.h>

typedef __attribute__((ext_vector_type(2))) float v2f;
typedef __attribute__((ext_vector_type(8))) float v8f;

#define B_  2
#define N_  8192
#define D_  512
#define H_  8
#define DK_ 64
#define KG_ 32

__device__ __forceinline__ v8f wmma_f32(v2f a, v2f b, v8f c) {
    // V_WMMA_F32_16X16X4_F32 : D(16x16 f32) = A(16x4 f32) * B(4x16 f32) + C
    return __builtin_amdgcn_wmma_f32_16x16x4_f32(
        /*neg_a=*/false, a, /*neg_b=*/false, b,
        /*c_mod=*/(short)0, c, /*reuse_a=*/false, /*reuse_b=*/false);
}

// ---------------------------------------------------------------------------
// Kernel 1: Y = x @ W^T for Wq/Wk/Wv, fused per-head LayerNorm (Q,K only),
// output layout (B, H, N, 64). One wave computes a 32(M) x 64(N) tile:
// two 16-row groups share the same four B fragments each k-step.
// ---------------------------------------------------------------------------
__global__ __launch_bounds__(256) void qkv_ln_kernel(
    const float* __restrict__ x,
    const float* __restrict__ Wq, const float* __restrict__ Wk,
    const float* __restrict__ Wv,
    float* __restrict__ Qln, float* __restrict__ Kln, float* __restrict__ Vout)
{
    const int wid  = blockIdx.x * 8 + (threadIdx.x >> 5);
    const int lane = threadIdx.x & 31;
    const int g    = lane >> 4;      // K-half selector for A/B fragments
    const int lm   = lane & 15;      // M (A) / N (B) index within fragment

    // 4096 wave-tiles per matrix: 512 M-tiles (32 rows) x 8 head-strips
    const int mat   = wid >> 12;
    const int rem   = wid & 4095;
    const int mtile = rem >> 3;
    const int h     = rem & 7;

    const float* W   = (mat == 0) ? Wq  : (mat == 1) ? Wk  : Wv;
    float*       dst = (mat == 0) ? Qln : (mat == 1) ? Kln : Vout;

    const int row0 = mtile * 32;
    const float* arow0 = x + (size_t)(row0 + lm) * D_ + 2 * g;
    const float* arow1 = arow0 + (size_t)16 * D_;
    const float* brow  = W + (size_t)(h * DK_ + lm) * D_ + 2 * g;

    v8f acc[2][4] = {};
    #pragma unroll 4
    for (int k0 = 0; k0 < D_; k0 += 4) {
        v2f a0 = *(const v2f*)(arow0 + k0);
        v2f a1 = *(const v2f*)(arow1 + k0);
        #pragma unroll
        for (int j = 0; j < 4; ++j) {
            v2f b = *(const v2f*)(brow + j * 16 * D_ + k0);
            acc[0][j] = wmma_f32(a0, b, acc[0][j]);
            acc[1][j] = wmma_f32(a1, b, acc[1][j]);
        }
    }

    // In-register LayerNorm over the 64 head channels (Q and K matrices only).
    // Row m = 16*t + v + 8*g lives in acc[t][0..3][v] across one half-wave.
    if (mat < 2) {
        #pragma unroll
        for (int t = 0; t < 2; ++t) {
            #pragma unroll
            for (int v = 0; v < 8; ++v) {
                float s  = acc[t][0][v] + acc[t][1][v] + acc[t][2][v] + acc[t][3][v];
                float s2 = acc[t][0][v]*acc[t][0][v] + acc[t][1][v]*acc[t][1][v]
                         + acc[t][2][v]*acc[t][2][v] + acc[t][3][v]*acc[t][3][v];
                #pragma unroll
                for (int off = 1; off < 16; off <<= 1) {  // stays within half-wave
                    s  += __shfl_xor(s,  off, 32);
                    s2 += __shfl_xor(s2, off, 32);
                }
                float mu  = s * (1.0f / 64.0f);
                float var = s2 * (1.0f / 64.0f) - mu * mu;
                float rs  = 1.0f / sqrtf(var + 1e-5f);
                acc[t][0][v] = (acc[t][0][v] - mu) * rs;
                acc[t][1][v] = (acc[t][1][v] - mu) * rs;
                acc[t][2][v] = (acc[t][2][v] - mu) * rs;
                acc[t][3][v] = (acc[t][3][v] - mu) * rs;
            }
        }
    }

    // Store to (B, H, N, 64)
    #pragma unroll
    for (int t = 0; t < 2; ++t) {
        #pragma unroll
        for (int v = 0; v < 8; ++v) {
            int r = row0 + 16 * t + v + 8 * g;
            int b = r >> 13;
            int n = r & (N_ - 1);
            size_t base = ((size_t)(b * H_ + h) * N_ + n) * DK_ + lm;
            dst[base +  0] = acc[t][0][v];
            dst[base + 16] = acc[t][1][v];
            dst[base + 32] = acc[t][2][v];
            dst[base + 48] = acc[t][3][v];
        }
    }
}

// ---------------------------------------------------------------------------
// Kernel 2: gathered attention. One wave per (b, h, n); 32 lanes = 32 keys.
// ---------------------------------------------------------------------------
__global__ __launch_bounds__(256) void attn_kernel(
    const float* __restrict__ Qln, const float* __restrict__ Kln,
    const float* __restrict__ V,   const int* __restrict__ idx,
    float* __restrict__ attn_out)
{
    const int wid  = blockIdx.x * 8 + (threadIdx.x >> 5);
    const int lane = threadIdx.x & 31;
    const int b    = wid >> 16;          // / (H*N)
    const int rem  = wid & 65535;
    const int h    = rem >> 13;
    const int n    = rem & (N_ - 1);

    const size_t head_base = (size_t)(b * H_ + h) * N_;
    const int kidx = idx[n * KG_ + lane];

    // scores: lane j = q . K[idx_j]
    const float* q    = Qln + (head_base + n) * DK_;
    const float* krow = Kln + (head_base + kidx) * DK_;
    float s = 0.0f;
    #pragma unroll
    for (int d0 = 0; d0 < DK_; d0 += 4) {
        float4 qv = *(const float4*)(q + d0);     // uniform address, cache hit
        float4 kv = *(const float4*)(krow + d0);
        s += qv.x * kv.x + qv.y * kv.y + qv.z * kv.z + qv.w * kv.w;
    }
    s *= 0.125f;  // 1/sqrt(64)

    // softmax across the 32 lanes
    float m = s;
    #pragma unroll
    for (int off = 16; off >= 1; off >>= 1) m = fmaxf(m, __shfl_xor(m, off, 32));
    float e = expf(s - m);
    float sum = e;
    #pragma unroll
    for (int off = 16; off >= 1; off >>= 1) sum += __shfl_xor(sum, off, 32);
    float w = e / sum;

    // out[d] = sum_j w_j * V[idx_j][d] ; lane owns 2 channels, coalesced rows
    float2 acc = make_float2(0.0f, 0.0f);
    #pragma unroll 8
    for (int j = 0; j < KG_; ++j) {
        float wj = __shfl(w, j, 32);
        int   kj = __shfl(kidx, j, 32);
        float2 vv = *(const float2*)(V + (head_base + kj) * DK_ + 2 * lane);
        acc.x += wj * vv.x;
        acc.y += wj * vv.y;
    }
    float* o = attn_out + ((size_t)(b * N_ + n)) * D_ + h * DK_ + 2 * lane;
    *(float2*)o = acc;
}

// ---------------------------------------------------------------------------
// Kernel 3: out = attn_out @ Wout^T + bout -> d_out (B, N, D).
// One wave computes a 32(M) x 64(N) tile.
// ---------------------------------------------------------------------------
__global__ __launch_bounds__(256) void out_proj_kernel(
    const float* __restrict__ A, const float* __restrict__ Wout,
    const float* __restrict__ bout, float* __restrict__ out)
{
    const int wid  = blockIdx.x * 8 + (threadIdx.x >> 5);
    const int lane = threadIdx.x & 31;
    const int g    = lane >> 4;
    const int lm   = lane & 15;
    const int mtile = wid >> 3;          // 0..511
    const int nt    = wid & 7;

    const int row0 = mtile * 32;
    const float* arow0 = A    + (size_t)(row0 + lm) * D_ + 2 * g;
    const float* arow1 = arow0 + (size_t)16 * D_;
    const float* brow  = Wout + (size_t)(nt * 64 + lm) * D_ + 2 * g;

    v8f acc[2][4] = {};
    #pragma unroll 4
    for (int k0 = 0; k0 < D_; k0 += 4) {
        v2f a0 = *(const v2f*)(arow0 + k0);
        v2f a1 = *(const v2f*)(arow1 + k0);
        #pragma unroll
        for (int j = 0; j < 4; ++j) {
            v2f b = *(const v2f*)(brow + j * 16 * D_ + k0);
            acc[0][j] = wmma_f32(a0, b, acc[0][j]);
            acc[1][j] = wmma_f32(a1, b, acc[1][j]);
        }
    }

    const float bb0 = bout[nt * 64      + lm];
    const float bb1 = bout[nt * 64 + 16 + lm];
    const float bb2 = bout[nt * 64 + 32 + lm];
    const float bb3 = bout[nt * 64 + 48 + lm];

    #pragma unroll
    for (int t = 0; t < 2; ++t) {
        #pragma unroll
        for (int v = 0; v < 8; ++v) {
            int r = row0 + 16 * t + v + 8 * g;
            float* o = out + (size_t)r * D_ + nt * 64 + lm;
            o[ 0] = acc[t][0][v] + bb0;
            o[16] = acc[t][1][v] + bb1;
            o[32] = acc[t][2][v] + bb2;
            o[48] = acc[t][3][v] + bb3;
        }
    }
}

// ---------------------------------------------------------------------------
extern "C" void kernel_launch(void* const* d_in, const int* in_sizes, int n_in,
                              void* d_out, int out_size, void* d_ws, size_t ws_size,
                              hipStream_t stream)
{
    const float* x    = (const float*)d_in[0];
    const int*   idx  = (const int*)  d_in[1];
    const float* Wq   = (const float*)d_in[2];
    const float* Wk   = (const float*)d_in[3];
    const float* Wv   = (const float*)d_in[4];
    const float* Wout = (const float*)d_in[5];
    const float* bout = (const float*)d_in[6];

    const size_t qkv_elems = (size_t)B_ * H_ * N_ * DK_;   // 8,388,608 floats
    float* Qln = (float*)d_ws;
    float* Kln = Qln + qkv_elems;
    float* Vws = Kln + qkv_elems;
    float* att = Vws + qkv_elems;

    // 3 matrices x 512 M-tiles x 8 head-strips = 12288 waves / 8 per block
    qkv_ln_kernel<<<1536, 256, 0, stream>>>(x, Wq, Wk, Wv, Qln, Kln, Vws);
    // B*H*N = 131072 waves / 8 per block
    attn_kernel<<<16384, 256, 0, stream>>>(Qln, Kln, Vws, idx, att);
    // 512 M-tiles x 8 strips = 4096 waves / 8 per block
    out_proj_kernel<<<512, 256, 0, stream>>>(att, Wout, bout, (float*)d_out);
}